// ConfidenceGate_22162031247910
// MI455X (gfx1250) — compile-verified
//
#include <hip/hip_runtime.h>
#include <math.h>

typedef __attribute__((ext_vector_type(2))) float v2f;
typedef __attribute__((ext_vector_type(8))) float v8f;
typedef __attribute__((ext_vector_type(4))) unsigned int v4u;
typedef __attribute__((ext_vector_type(8))) int v8i;
typedef __attribute__((ext_vector_type(4))) int v4i;

#define M_ROWS   512
#define C_CH     256
#define GG       49
#define P_DIM    32
#define ROW_F    (C_CH * GG)          // 12544 floats per (m) row
#define ROW_B    (ROW_F * 4)          // 50176 bytes per row
#define NROWS    1024                 // 512 curr + 512 prev

__device__ __forceinline__ float wave_sum32(float v) {
    #pragma unroll
    for (int off = 16; off > 0; off >>= 1)
        v += __shfl_xor(v, off, 32);
    return v;
}

// ---------------------------------------------------------------------------
// Kernel A: GxG mean-pool of x[0] and prev_x[0] -> pool[1024][256]
// rows 0..511 = x_pool, rows 512..1023 = prev_pool.
// The 50 KB contiguous row is DMA'd into LDS by the Tensor Data Mover
// (tensor_load_to_lds, tracked by TENSORcnt), then 256 threads reduce
// their 49-element channel windows.
// D# layout per CDNA5 ISA (1-row 2D tile).
// ---------------------------------------------------------------------------
__global__ void cg_pool_kernel(const float* __restrict__ x,
                               const float* __restrict__ px,
                               float* __restrict__ pool) {
    __shared__ float tile[ROW_F];     // 50176 B of LDS
    const int r = blockIdx.x;         // 0..1023
    const float* src = (r < M_ROWS) ? (x + (size_t)r * ROW_F)
                                    : (px + (size_t)(r - M_ROWS) * ROW_F);

    if (threadIdx.x < 32) {           // wave 0 issues the TDM descriptor
        const unsigned long long gaddr = (unsigned long long)(uintptr_t)src;
        // flat LDS aperture: addr[31:0] carries the workgroup-relative offset
        const unsigned int lds_addr = (unsigned int)(uintptr_t)(void*)&tile[0];

        // D# group 0: count=1, lds_addr, global_addr[56:0], type=2
        v4u g0;
        g0[0] = 1u;                                   // count=1, is_restore=0
        g0[1] = lds_addr;                             // bits 63:32  lds_addr
        g0[2] = (unsigned int)(gaddr & 0xffffffffu);  // bits 95:64  gaddr lo
        g0[3] = ((unsigned int)(gaddr >> 32) & 0x01ffffffu) // bits 120:96
                | 0x80000000u;                        // bits 127:126 type=2

        // D# group 1: data_size=2 (4B), dims/strides for a 12544x1 tile
        v8i g1;
        g1[0] = (int)(2u << 16);                      // data_size=4B, mask=0
        g1[1] = (int)((ROW_F & 0xffffu) << 16);       // tensor_dim0[15:0]
        g1[2] = (int)(((ROW_F >> 16) & 0xffffu)       // tensor_dim0[31:16]
                | (1u << 16));                        // tensor_dim1 = 1
        g1[3] = (int)((ROW_F & 0xffffu) << 16);       // tile_dim0 = 12544
        g1[4] = 1;                                    // tile_dim1=1, tile_dim2=0
        g1[5] = ROW_F;                                // tensor_dim0_stride lo
        g1[6] = 0;                                    // stride hi, dim1_stride lo
        g1[7] = 0;                                    // dim1_stride hi

        v4i g2 = {};                                  // <=2D: groups 2/3 unused
        v4i g3 = {};

#if defined(__clang_major__) && (__clang_major__ >= 23)
        v8i g4 = {};
        __builtin_amdgcn_tensor_load_to_lds(g0, g1, g2, g3, g4, 0);
#else
        __builtin_amdgcn_tensor_load_to_lds(g0, g1, g2, g3, 0);
#endif
        __builtin_amdgcn_s_wait_tensorcnt(0);
    }
    __syncthreads();

    const int c = threadIdx.x;        // 0..255
    float s = 0.0f;
    #pragma unroll
    for (int j = 0; j < GG; ++j)
        s += tile[c * GG + j];
    pool[r * C_CH + c] = s * (1.0f / 49.0f);
}

// ---------------------------------------------------------------------------
// Kernel B: per-m match statistics (batch 0). One wave per m.
// stats[m*8 + {0:p_dummy,1:p_max,2:p_gap,3:ent,4:real_mass,5:top1}]
// ---------------------------------------------------------------------------
__global__ void cg_match_kernel(const float* __restrict__ match,
                                float* __restrict__ stats) {
    const int wave = threadIdx.x >> 5;
    const int lane = threadIdx.x & 31;
    const int m = blockIdx.x * 8 + wave;
    const float* row = match + (size_t)m * 513;

    float sum = 0.0f, plogp = 0.0f;
    float m1 = -3.0e38f, m2 = -3.0e38f;
    int   i1 = 0x7fffffff;
    for (int j = lane; j < 512; j += 32) {
        float v = row[j];
        sum += v;
        float rr = fmaxf(v, 1e-9f);
        plogp += rr * logf(rr);
        if (v > m1 || (v == m1 && j < i1)) { m2 = m1; m1 = v; i1 = j; }
        else                               { m2 = fmaxf(m2, v); }
    }
    #pragma unroll
    for (int off = 16; off > 0; off >>= 1) {
        sum   += __shfl_xor(sum,   off, 32);
        plogp += __shfl_xor(plogp, off, 32);
        float om1 = __shfl_xor(m1, off, 32);
        int   oi1 = __shfl_xor(i1, off, 32);
        float om2 = __shfl_xor(m2, off, 32);
        if (om1 > m1 || (om1 == m1 && oi1 < i1)) {
            m2 = fmaxf(m1, om2); m1 = om1; i1 = oi1;
        } else {
            m2 = fmaxf(m2, om1);
        }
    }
    if (lane == 0) {
        float* st = stats + m * 8;
        bool has_real = sum > 1e-9f;
        st[0] = row[512];                       // p_dummy
        st[1] = m1;                             // p_max
        st[2] = m1 - m2;                        // p_gap
        st[3] = -plogp;                         // entropy
        st[4] = sum;                            // real_mass
        st[5] = (float)(has_real ? i1 : 0);     // top1
    }
}

// ---------------------------------------------------------------------------
// Kernel C: V = pool(1024x256) @ proj_w^T(256x32) + proj_b via f32 WMMA.
// One wave per 16-row M-tile; two N-halves (cols 0-15, 16-31); K stepped by 4.
// f32 WMMA VGPR layouts per CDNA5 ISA.
// ---------------------------------------------------------------------------
__global__ void cg_gemm_kernel(const float* __restrict__ pool,
                               const float* __restrict__ proj_w,
                               const float* __restrict__ proj_b,
                               float* __restrict__ V) {
    const int m0   = blockIdx.x * 16;
    const int lane = threadIdx.x;        // 32 threads, 1 wave
    const int lo   = lane & 15;
    const int hi   = lane >> 4;          // 0 or 1 -> K-half / M-half selector

    const float* arow  = pool   + (size_t)(m0 + lo) * C_CH + 2 * hi;
    const float* b0row = proj_w + (size_t)lo        * C_CH + 2 * hi;
    const float* b1row = proj_w + (size_t)(lo + 16) * C_CH + 2 * hi;

    v8f acc0 = {};
    v8f acc1 = {};
    #pragma unroll 4
    for (int k = 0; k < C_CH; k += 4) {
        v2f a  = *(const v2f*)(arow  + k);   // A[M=lo, K=k+2*hi .. +1]
        v2f b0 = *(const v2f*)(b0row + k);   // B[K, N=lo]
        v2f b1 = *(const v2f*)(b1row + k);   // B[K, N=lo+16]
        acc0 = __builtin_amdgcn_wmma_f32_16x16x4_f32(
            false, a, false, b0, (short)0, acc0, false, false);
        acc1 = __builtin_amdgcn_wmma_f32_16x16x4_f32(
            false, a, false, b1, (short)0, acc1, false, false);
    }
    const float bias0 = proj_b[lo];
    const float bias1 = proj_b[lo + 16];
    #pragma unroll
    for (int r = 0; r < 8; ++r) {
        int row = m0 + hi * 8 + r;           // D: vgpr r, lane half -> M
        V[row * P_DIM + lo]      = acc0[r] + bias0;
        V[row * P_DIM + lo + 16] = acc1[r] + bias1;
    }
}

// ---------------------------------------------------------------------------
// Kernel D: per-row LayerNorm(32) then l2-normalize, in place. Wave per row.
// ---------------------------------------------------------------------------
__global__ void cg_ln_kernel(float* __restrict__ V,
                             const float* __restrict__ ln_g,
                             const float* __restrict__ ln_b) {
    const int wave = threadIdx.x >> 5;
    const int lane = threadIdx.x & 31;
    const int row = blockIdx.x * 8 + wave;

    float v   = V[row * P_DIM + lane];
    float mu  = wave_sum32(v) * (1.0f / 32.0f);
    float d   = v - mu;
    float var = wave_sum32(d * d) * (1.0f / 32.0f);
    float y   = ln_g[lane] * d / sqrtf(var + 1e-5f) + ln_b[lane];
    float n   = sqrtf(wave_sum32(y * y));
    V[row * P_DIM + lane] = y / fmaxf(n, 1e-12f);
}

// ---------------------------------------------------------------------------
// Kernel E: cos-sim + 5->32->1 gate MLP + sigmoid + masks. Wave per m.
// ---------------------------------------------------------------------------
__global__ void cg_gate_kernel(const float* __restrict__ U,
                               const float* __restrict__ stats,
                               const float* __restrict__ w1,
                               const float* __restrict__ b1,
                               const float* __restrict__ w2,
                               const float* __restrict__ b2,
                               float* __restrict__ out) {
    const int wave = threadIdx.x >> 5;
    const int lane = threadIdx.x & 31;
    const int m = blockIdx.x * 8 + wave;

    const float* st = stats + m * 8;
    const float p_dummy   = st[0];
    const float p_max     = st[1];
    const float p_gap     = st[2];
    const float ent       = st[3];
    const float real_mass = st[4];
    const int   top1      = (int)st[5];

    float c = U[m * P_DIM + lane] * U[(M_ROWS + top1) * P_DIM + lane];
    float cos = wave_sum32(c);
    if (!(real_mass > 1e-9f)) cos = 0.0f;

    const float f0 = 1.0f - p_dummy;
    const float f1 = p_max;
    const float f2 = p_gap;
    const float f3 = -ent;
    const float f4 = cos;

    const float* wr = w1 + lane * 5;
    float h = fmaf(f0, wr[0],
              fmaf(f1, wr[1],
              fmaf(f2, wr[2],
              fmaf(f3, wr[3],
              fmaf(f4, wr[4], b1[lane])))));
    h = fmaxf(h, 0.0f);

    float logit = wave_sum32(h * w2[lane]) + b2[0];
    float cc = 1.0f / (1.0f + expf(-logit));
    if (real_mass <= 1e-6f) cc = 0.0f;
    cc = fminf(fmaxf(cc, 0.001f), 0.999f);
    if (lane == 0) out[m] = cc;
}

// ---------------------------------------------------------------------------
extern "C" void kernel_launch(void* const* d_in, const int* in_sizes, int n_in,
                              void* d_out, int out_size, void* d_ws, size_t ws_size,
                              hipStream_t stream) {
    const float* x      = (const float*)d_in[0];
    const float* px     = (const float*)d_in[1];
    const float* match  = (const float*)d_in[2];
    const float* proj_w = (const float*)d_in[3];
    const float* proj_b = (const float*)d_in[4];
    const float* ln_g   = (const float*)d_in[5];
    const float* ln_b   = (const float*)d_in[6];
    const float* w1     = (const float*)d_in[7];
    const float* b1     = (const float*)d_in[8];
    const float* w2     = (const float*)d_in[9];
    const float* b2     = (const float*)d_in[10];

    float* ws    = (float*)d_ws;
    float* pool  = ws;                              // 1024*256 floats
    float* V     = pool + NROWS * C_CH;             // 1024*32 floats
    float* stats = V + NROWS * P_DIM;               // 512*8 floats
    float* out   = (float*)d_out;                   // 512 floats

    cg_pool_kernel <<<NROWS,        256, 0, stream>>>(x, px, pool);
    cg_match_kernel<<<M_ROWS / 8,   256, 0, stream>>>(match, stats);
    cg_gemm_kernel <<<NROWS / 16,    32, 0, stream>>>(pool, proj_w, proj_b, V);
    cg_ln_kernel   <<<NROWS / 8,    256, 0, stream>>>(V, ln_g, ln_b);
    cg_gate_kernel <<<M_ROWS / 8,   256, 0, stream>>>(V, stats, w1, b1, w2, b2, out);
}